// _MSSIMLossDebug_18691697672271
// MI455X (gfx1250) — compile-verified
//
#include <hip/hip_runtime.h>

typedef float v2f __attribute__((ext_vector_type(2)));
typedef float v8f __attribute__((ext_vector_type(8)));

namespace {
constexpr int Bn = 2, Cn = 4, Dn = 64, Hn = 160, Wn = 160;
constexpr int T = 16;       // output tile edge
constexpr int HALO = 5;     // 11-tap Gaussian -> 5 halo
constexpr int S = 26;       // T + 2*HALO
constexpr int SP = 28;      // halo padded to multiple of 4 for K=4 WMMA chunks
// Bank-conflict-free padded strides (lane-half address delta == 16 banks mod 64):
constexpr int RS1 = 18;     // st1 row (w) stride
constexpr int RS2 = 18;     // st2 row (h) stride
constexpr int PS2 = 296;    // st2 plane (d) stride  (16*18=288 -> +8)
constexpr int PS5 = 258;    // out5 plane (d) stride (16*16=256 -> +2)
constexpr int FS5 = 16 * PS5; // out5 per-field stride
constexpr float C1c = 1e-4f, C2c = 9e-4f;
constexpr float INV_COUNT = 1.0f / (float)((long)Bn * Dn * Hn * Wn);
}

// Banded Gaussian conv-matrix entry: G[j][o] = g[j-o], taps t in [0,10], sigma=1.5
__device__ __forceinline__ float gwf(int j, int o) {
  int t = j - o;
  if ((unsigned)t > 10u) return 0.0f;
  float d = (float)(t - 5);
  return __expf(d * d * (-1.0f / 4.5f)) * 0.26601172f; // 1/sum of unnormalized taps
}

// Load 28 halo samples of one row from both images, form field F on the fly,
// and run the 7-step K=4 f32 WMMA chain. Branch-free (F is compile-time).
template <int F>
__device__ __forceinline__ v8f conv_w_mac(const float* __restrict__ r1,
                                          const float* __restrict__ r2,
                                          const v2f* __restrict__ gfrag,
                                          int kbase) {
  v2f a[7];
#pragma unroll
  for (int kc = 0; kc < 7; ++kc) {            // batch loads first (LDS clause)
    int j0 = kc * 4 + kbase;
    v2f x1 = *(const v2f*)(r1 + j0);
    v2f x2 = *(const v2f*)(r2 + j0);
    if      (F == 0) a[kc] = x1;
    else if (F == 1) a[kc] = x2;
    else if (F == 2) a[kc] = x1 * x1;
    else if (F == 3) a[kc] = x2 * x2;
    else             a[kc] = x1 * x2;
  }
  v8f acc = {};
#pragma unroll
  for (int kc = 0; kc < 7; ++kc)              // back-to-back WMMA chain
    acc = __builtin_amdgcn_wmma_f32_16x16x4_f32(false, a[kc], false, gfrag[kc],
                                                (short)0, acc, false, false);
  return acc;
}

// W-axis conv over flattened (d,h) rows. Rows 0..671 (groups 0..41) are fully
// valid -> no store guards, no div per store. Group 42 handled as uniform tail.
template <int F>
__device__ __forceinline__ void conv_w_pass(const float* __restrict__ in1,
                                            const float* __restrict__ in2,
                                            float* __restrict__ st1,
                                            const v2f* __restrict__ gfrag,
                                            int wave, int nlo, int kbase, int mAdd) {
  for (int grp = wave; grp < 42; grp += 8) {
    int row = grp * 16 + nlo;                 // (d*S+h)*SP == row*SP
    v8f acc = conv_w_mac<F>(&in1[row * SP], &in2[row * SP], gfrag, kbase);
    // st1 addr identity: ((orow/26)*28 + orow%26)*18 + nlo
    //                  = orow*18 + (orow/26)*36 + nlo, with at most one wrap over r.
    int b0 = grp * 16 + mAdd;
    int od0 = b0 / 26;
    int oh0 = b0 - od0 * 26;
    int addrBase = b0 * RS1 + od0 * 36 + nlo;
#pragma unroll
    for (int r = 0; r < 8; ++r) {
      int adj = (oh0 + r >= 26) ? 36 : 0;
      st1[addrBase + r * RS1 + adj] = acc[r];
    }
  }
  if (wave == 2) {                            // group 42 (42 % 8 == 2), uniform
    int row = 672 + nlo;                      // rows 672..687; >=676 clamped
    int rowc = row < S * S ? row : 0;
    v8f acc = conv_w_mac<F>(&in1[rowc * SP], &in2[rowc * SP], gfrag, kbase);
    if (mAdd == 0) {                          // only rows 672..675 are real
#pragma unroll
      for (int r = 0; r < 4; ++r)             // od=25, oh=22+r -> 25*28+22 = 722
        st1[(722 + r) * RS1 + nlo] = acc[r];
    }
  }
}

__global__ __launch_bounds__(256, 1) void mssim3d_fused_kernel(
    const float* __restrict__ img1, const float* __restrict__ img2,
    float* __restrict__ ws) {
  // ~313 KB LDS: CDNA5-only (320 KB per WGP)
  __shared__ __align__(16) float in1[S * S * SP];   // img1 halo tile [d][h][w pad28]
  __shared__ __align__(16) float in2[S * S * SP];   // img2 halo tile
  __shared__ __align__(16) float st1[S * SP * RS1]; // after W conv [d][h pad28][w]
  __shared__ __align__(16) float st2[SP * PS2];     // after H conv [d pad28][h][w]
  __shared__ __align__(16) float out5[5 * FS5];     // 5 conv fields, 16^3 (padded)
  __shared__ float red[256];

  const int tid  = threadIdx.x;
  const int lane = tid & 31;          // wave32 (gfx1250)
  const int wave = tid >> 5;          // 8 waves / block
  const int nlo  = lane & 15;         // M(A) / N(B,C) index
  const int kbase = (lane >> 4) << 1; // K pair base in f32 WMMA operand layout
  const int mAdd  = (lane >> 4) << 3; // C layout: lanes 16-31 hold M+8

  const int wt = blockIdx.x, ht = blockIdx.y, bz = blockIdx.z;
  const int dt = bz & 3, bc = bz >> 2;
  const int ch = bc & 3, bi = bc >> 2;
  const int d0 = dt * T, h0 = ht * T, w0 = wt * T;

  // Constant Gaussian fragments; identical per-lane layout serves as
  // B operand (W pass) and A operand (H/D passes).
  v2f gfrag[7];
#pragma unroll
  for (int kc = 0; kc < 7; ++kc) {
    int j0 = kc * 4 + kbase;
    gfrag[kc].x = gwf(j0, nlo);
    gfrag[kc].y = gwf(j0 + 1, nlo);
  }

  // ---- Stage 0: global -> LDS halo load, zero SAME-padding ----
  const size_t chanBase = (size_t)((bi * Cn + ch) * Dn) * (size_t)(Hn * Wn);
  for (int li = tid; li < S * S * SP; li += 256) {
    int lw = li % SP;
    int lh = (li / SP) % S;
    int ld = li / (SP * S);
    int gd = d0 + ld - HALO, gh = h0 + lh - HALO, gw = w0 + lw - HALO;
    float v1 = 0.f, v2 = 0.f;
    if (lw < S && (unsigned)gd < (unsigned)Dn && (unsigned)gh < (unsigned)Hn &&
        (unsigned)gw < (unsigned)Wn) {
      size_t idx = chanBase + (size_t)(gd * Hn + gh) * Wn + gw;
      v1 = img1[idx];
      v2 = img2[idx];
    }
    in1[li] = v1;
    in2[li] = v2;
  }
  // Zero padded rows/planes once (stores in the passes never touch them).
  for (int li = tid; li < S * 2 * T; li += 256) {
    int d = li >> 5, r = (li >> 4) & 1, w = li & 15;
    st1[(d * SP + S + r) * RS1 + w] = 0.f;
  }
  for (int li = tid; li < 2 * PS2; li += 256) st2[S * PS2 + li] = 0.f;
  __syncthreads();

  // ---- 5 fields x separable 3-axis conv; each 1-D conv = banded f32 WMMA ----
  for (int f = 0; f < 5; ++f) {
    switch (f) {  // block-uniform dispatch -> straight-line W pass per field
      case 0: conv_w_pass<0>(in1, in2, st1, gfrag, wave, nlo, kbase, mAdd); break;
      case 1: conv_w_pass<1>(in1, in2, st1, gfrag, wave, nlo, kbase, mAdd); break;
      case 2: conv_w_pass<2>(in1, in2, st1, gfrag, wave, nlo, kbase, mAdd); break;
      case 3: conv_w_pass<3>(in1, in2, st1, gfrag, wave, nlo, kbase, mAdd); break;
      default: conv_w_pass<4>(in1, in2, st1, gfrag, wave, nlo, kbase, mAdd); break;
    }
    __syncthreads();

    // H-axis: one 16x16 tile per d-plane (26 planes)
    for (int dp = wave; dp < S; dp += 8) {
      const float* base = &st1[dp * SP * RS1];
      v2f bb[7];
#pragma unroll
      for (int kc = 0; kc < 7; ++kc) {
        int j0 = kc * 4 + kbase;
        bb[kc].x = base[j0 * RS1 + nlo];
        bb[kc].y = base[(j0 + 1) * RS1 + nlo];
      }
      v8f acc = {};
#pragma unroll
      for (int kc = 0; kc < 7; ++kc)
        acc = __builtin_amdgcn_wmma_f32_16x16x4_f32(false, gfrag[kc], false, bb[kc],
                                                    (short)0, acc, false, false);
      int sbase = dp * PS2 + mAdd * RS2 + nlo;
#pragma unroll
      for (int r = 0; r < 8; ++r)
        st2[sbase + r * RS2] = acc[r];
    }
    __syncthreads();

    // D-axis: one 16x16 tile per h-plane (16 planes)
    for (int hp = wave; hp < T; hp += 8) {
      v2f bb[7];
#pragma unroll
      for (int kc = 0; kc < 7; ++kc) {
        int j0 = kc * 4 + kbase;
        bb[kc].x = st2[j0 * PS2 + hp * RS2 + nlo];
        bb[kc].y = st2[(j0 + 1) * PS2 + hp * RS2 + nlo];
      }
      v8f acc = {};
#pragma unroll
      for (int kc = 0; kc < 7; ++kc)
        acc = __builtin_amdgcn_wmma_f32_16x16x4_f32(false, gfrag[kc], false, bb[kc],
                                                    (short)0, acc, false, false);
      int obase = f * FS5 + mAdd * PS5 + hp * 16 + nlo;
#pragma unroll
      for (int r = 0; r < 8; ++r)
        out5[obase + r * PS5] = acc[r];
    }
    __syncthreads();
  }

  // ---- SSIM map + tile reduction ----
  const int hh = (tid >> 4) & 15, ww = tid & 15;
  float part = 0.f;
#pragma unroll
  for (int k = 0; k < 16; ++k) {
    int off = k * PS5 + hh * 16 + ww;
    float mu1 = out5[off];
    float mu2 = out5[FS5 + off];
    float e11 = out5[2 * FS5 + off];
    float e22 = out5[3 * FS5 + off];
    float e12 = out5[4 * FS5 + off];
    float mu1s = mu1 * mu1, mu2s = mu2 * mu2, m12 = mu1 * mu2;
    float s1q = e11 - mu1s, s2q = e22 - mu2s, s12 = e12 - m12;
    float num = (2.f * m12 + C1c) * (2.f * s12 + C2c);
    float den = (mu1s + mu2s + C1c) * (s1q + s2q + C2c);
    part += num / den;
  }
  red[tid] = part;
  __syncthreads();
  for (int s = 128; s > 0; s >>= 1) {
    if (tid < s) red[tid] += red[tid + s];
    __syncthreads();
  }
  if (tid == 0) atomicAdd(&ws[ch], red[0]);
}

__global__ void mssim3d_init_kernel(float* __restrict__ ws) {
  if (threadIdx.x < 4) ws[threadIdx.x] = 0.f;
}

__global__ void mssim3d_final_kernel(const float* __restrict__ ws,
                                     float* __restrict__ out) {
  if (threadIdx.x < 4) out[threadIdx.x] = 1.0f - ws[threadIdx.x] * INV_COUNT;
}

extern "C" void kernel_launch(void* const* d_in, const int* in_sizes, int n_in,
                              void* d_out, int out_size, void* d_ws, size_t ws_size,
                              hipStream_t stream) {
  (void)in_sizes; (void)n_in; (void)out_size; (void)ws_size;
  const float* img1 = (const float*)d_in[0];
  const float* img2 = (const float*)d_in[1];
  float* out = (float*)d_out;
  float* ws  = (float*)d_ws;

  mssim3d_init_kernel<<<1, 32, 0, stream>>>(ws);
  dim3 grid(Wn / T, Hn / T, (Dn / T) * Bn * Cn);  // 10 x 10 x 32 = 3200 tiles
  mssim3d_fused_kernel<<<grid, 256, 0, stream>>>(img1, img2, ws);
  mssim3d_final_kernel<<<1, 32, 0, stream>>>(ws, out);
}